// ASIC_87007447483060
// MI455X (gfx1250) — compile-verified
//
#include <hip/hip_runtime.h>
#include <math.h>

#ifndef __has_builtin
#define __has_builtin(x) 0
#endif

// Paths, preferred order: TDM tensor_load_to_lds > per-lane async > plain.
#if __has_builtin(__builtin_amdgcn_tensor_load_to_lds)
#define USE_TDM 1
#else
#define USE_TDM 0
#endif
#if __has_builtin(__builtin_amdgcn_global_load_async_to_lds_b32)
#define ASYNC_LDS 1
#else
#define ASYNC_LDS 0
#endif

typedef __attribute__((address_space(1))) int g_int;
typedef __attribute__((address_space(3))) int l_int;
typedef unsigned int v4u __attribute__((ext_vector_type(4)));
typedef int v8i __attribute__((ext_vector_type(8)));
typedef int v4i __attribute__((ext_vector_type(4)));

__device__ __forceinline__ void async_copy_b32(const float* g, float* l) {
#if ASYNC_LDS
  __builtin_amdgcn_global_load_async_to_lds_b32((g_int*)g, (l_int*)l, 0, 0);
#else
  *l = *g;
#endif
}

__device__ __forceinline__ void wait_async_zero() {
#if ASYNC_LDS
#if __has_builtin(__builtin_amdgcn_s_wait_asynccnt)
  __builtin_amdgcn_s_wait_asynccnt(0);
#else
  asm volatile("s_wait_asynccnt 0" ::: "memory");
#endif
#endif
}

__device__ __forceinline__ float fast_sigmoid(float g) {
  return 1.0f / (1.0f + __expf(-g));
}

#define TILE 256

// Border ring of each rail plane (non-scattered positions), including the
// x-write into plane (1,1), column 0, rows [0, n).
__global__ void asic_border(const float* __restrict__ x,
                            const float* __restrict__ rail,
                            const int* __restrict__ mask,
                            const float* __restrict__ tg,
                            float* __restrict__ out, int n) {
  const int t = blockIdx.x * blockDim.x + threadIdx.x;
  const int per = 2 * n + 1;
  if (t >= 4 * per) return;
  const int q = t / per;  // plane 0..3
  const int j = t % per;
  int R, C;
  if (q < 2) {  // planes (0,0),(0,1): ring = row n + col n
    if (j <= n) { R = n; C = j; } else { R = j - (n + 1); C = n; }
  } else {      // planes (1,0),(1,1): ring = row 0 + col 0
    if (j <= n) { R = 0; C = j; } else { R = j - n; C = 0; }
  }
  const int S = n + 1;
  const int flat = q * S * S + R * S + C;
  const float v = (q == 3 && C == 0 && R < n) ? x[R] : rail[flat];
  out[flat] = v * (float)mask[flat] * tg[0];
}

__global__ __launch_bounds__(TILE) void asic_interior(
    const float* __restrict__ x,
    const float* __restrict__ tg,
    const float* __restrict__ rail,
    const int* __restrict__ mask,
    float* __restrict__ out, int n) {
  __shared__ float tgs[32 * TILE];  // 32 KiB tile: 32 streams x 256 cols
  const int tid = threadIdx.x;
  const int chunks = n / TILE;          // n = 1024 -> 4
  const int r = blockIdx.x / chunks;    // row
  const int c0 = (blockIdx.x % chunks) * TILE;
  const int c = c0 + tid;

  const int nn = n * n;

#if USE_TDM
  // One Tensor Data Mover op per block: 2D tile, 32 rows (stride n*n elems)
  // x 256 contiguous f32 -> LDS, laid out row-major (tgs[k*TILE + x]).
  if (tid < 32) {  // wave 0 issues the TDM descriptor
    const unsigned lds_base = (unsigned)(unsigned long long)(l_int*)&tgs[0];
    const unsigned long long ga =
        (unsigned long long)(const void*)(tg + r * n + c0);
    // D# group 0: count=1 | lds_addr | global_addr[56:0] | type=2
    v4u g0 = {1u, lds_base, (unsigned)(ga & 0xFFFFFFFFu),
              (unsigned)((ga >> 32) & 0x01FFFFFFu) | 0x80000000u};
    // D# group 1: data_size=4B; tensor_dim0/1 = 0x7FFFFFFF (no OOB clip);
    // tile_dim0=256, tile_dim1=32, tile_dim2=0; stride0=n*n elements.
    v8i g1 = {(int)0x00020000u,
              (int)0xFFFF0000u,                  // dim0[15:0]<<16
              (int)0xFFFF7FFFu,                  // dim0[31:16] | dim1[15:0]<<16
              (int)((TILE << 16) | 0x7FFF),      // dim1[31:16] | tile_dim0<<16
              32,                                // tile_dim1 | tile_dim2<<16
              (int)(unsigned)nn,                 // stride0[31:0]
              0, 0};
    v4i g2 = {0, 0, 0, 0};
    v4i g3 = {0, 0, 0, 0};
    v8i g4 = {0, 0, 0, 0, 0, 0, 0, 0};  // extra group (6-arg builtin form)
    __builtin_amdgcn_tensor_load_to_lds(g0, g1, g2, g3, g4, 0);
  }
#else
#pragma unroll
  for (int k = 0; k < 32; ++k) {
    async_copy_b32(tg + k * nn + r * n + c, &tgs[k * TILE + tid]);
  }
#endif

  // Streaming-read hint for the next column chunk (global_prefetch_b8).
  __builtin_prefetch(tg + r * n + c + TILE, 0, 0);

  // Rail gathers overlap with the tile DMA in flight.
  const int S = n + 1;
  const int P = S * S;
  const int rc1 = (r + 1) * S + (c + 1);
  const int rc0 = r * S + c;

  float ni[4];
  ni[0] = rail[0 * P + rc1];
  ni[1] = rail[1 * P + rc1];
  ni[2] = rail[2 * P + rc0];
  ni[3] = (c == 0) ? x[r] : rail[3 * P + rc0];

  const float tg0 = tg[0];

  float f0[4], f1[4];
#pragma unroll
  for (int j = 0; j < 4; ++j) {
    f1[j] = ni[j];
    f0[j] = 1.0f - ni[j];
  }

  const int oflat[4] = {0 * P + rc0, 1 * P + rc0, 2 * P + rc1, 3 * P + rc1};
  const int oth[4][3] = {{1, 2, 3}, {0, 2, 3}, {0, 1, 3}, {0, 1, 2}};

#if USE_TDM
  if (tid < 32) {
#if __has_builtin(__builtin_amdgcn_s_wait_tensorcnt)
    __builtin_amdgcn_s_wait_tensorcnt(0);
#else
    asm volatile("s_wait_tensorcnt 0" ::: "memory");
#endif
  }
#else
  wait_async_zero();
#endif
  __syncthreads();

#pragma unroll
  for (int i = 0; i < 4; ++i) {
    const int o0 = oth[i][0], o1 = oth[i][1], o2 = oth[i][2];
    const float a = ni[i];
    float acc = 0.0f;
#pragma unroll
    for (int p = 0; p < 8; ++p) {
      // soft-AND weight for bit pattern p over the leave-one-out inputs
      const float w = ((p & 4) ? f1[o0] : f0[o0]) *
                      ((p & 2) ? f1[o1] : f0[o1]) *
                      ((p & 1) ? f1[o2] : f0[o2]);
      const float tw = fast_sigmoid(tgs[(i * 8 + p) * TILE + tid]);
      acc += (tw * a + (1.0f - tw) * (1.0f - a)) * w;
    }
    acc = fminf(fmaxf(acc, 0.0f), 1.0f);
    const int of = oflat[i];
    out[of] = acc * (float)mask[of] * tg0;
  }
}

extern "C" void kernel_launch(void* const* d_in, const int* in_sizes, int n_in,
                              void* d_out, int out_size, void* d_ws, size_t ws_size,
                              hipStream_t stream) {
  const float* x    = (const float*)d_in[0];
  const float* tg   = (const float*)d_in[1];
  const float* rail = (const float*)d_in[2];
  const int*   mask = (const int*)d_in[3];
  float* out = (float*)d_out;
  const int n = in_sizes[0];  // 1024

  const int btot = 4 * (2 * n + 1);
  asic_border<<<(btot + 255) / 256, 256, 0, stream>>>(x, rail, mask, tg, out, n);

  const int chunks = n / TILE;
  asic_interior<<<dim3(n * chunks), dim3(TILE), 0, stream>>>(x, tg, rail, mask,
                                                             out, n);
}